// LaneATT_21440476742263
// MI455X (gfx1250) — compile-verified
//
#include <hip/hip_runtime.h>
#include <hip/hip_bf16.h>
#include <math.h>

typedef __attribute__((ext_vector_type(2))) float v2f;
typedef __attribute__((ext_vector_type(8))) float v8f;
typedef __attribute__((ext_vector_type(4))) int   v4i;

#define IMG_W 640.0f
#define IMG_H 360.0f
#define FH 11
#define FW 20
#define CF 64
#define CB 512
#define NPROP 4
#define FDIM 704   /* CF*FH */
#define HW 220     /* FH*FW */
#define NPAD 224   /* HW padded to multiple of 16 */
#define KS 16      /* K slab (512/16 = 32 slabs) */
#define APAD 20    /* A slab row pad: 16B-aligned rows, conflict-free reads */

#ifndef __has_builtin
#define __has_builtin(x) 0
#endif

#if __has_builtin(__builtin_amdgcn_global_load_async_to_lds_b128)
#define HAVE_ASYNC 1
typedef __attribute__((address_space(1))) v4i* gv4p;   // global int4*
typedef __attribute__((address_space(3))) v4i* lv4p;   // LDS int4*
#else
#define HAVE_ASYNC 0
#endif

#if __has_builtin(__builtin_amdgcn_s_wait_asynccnt)
#define WAIT_ASYNC() __builtin_amdgcn_s_wait_asynccnt(0)
#elif HAVE_ASYNC
#define WAIT_ASYNC() asm volatile("s_wait_asynccnt 0" ::: "memory")
#else
#define WAIT_ASYNC() ((void)0)
#endif

// ---------------------------------------------------------------------------
// Kernel 1: per-batch 1x1 conv as WMMA GEMM:
//   f[b] (64x220) = conv_w (64x512) @ feat[b] (512x220) + conv_b
// 1 block/batch, 256 thr = 8 wave32s, 7 of 56 16x16 tiles per wave.
// Double-buffered LDS slabs fed by async global->LDS (ASYNCcnt) when available.
// ---------------------------------------------------------------------------
__global__ __launch_bounds__(256) void laneatt_conv_wmma(
    const float* __restrict__ feat, const float* __restrict__ conv_w,
    const float* __restrict__ conv_b, float* __restrict__ fout)
{
  __shared__ float sB[2][KS * NPAD];   // feat slabs  [16][224]  (28.0 KB)
  __shared__ float sA[2][64 * APAD];   // conv_w slabs [64][20]  (10.0 KB)

  const int b   = blockIdx.x;
  const int tid = threadIdx.x;
  const int wv  = tid >> 5;     // wave id 0..7
  const int ln  = tid & 31;
  const int lr  = ln & 15;      // row/col within 16-tile
  const int lh  = ln >> 4;      // half selector (K pair / M+8)

  const float* featb = feat + (size_t)b * (CB * HW);

  // zero-fill the 4 padded columns of each B-slab row once (never overwritten)
  if (tid < 2 * KS * 4) {
    int bufi = tid >> 6;             // 0..1
    int r    = (tid >> 2) & (KS-1);  // row
    int c    = tid & 3;
    sB[bufi][r * NPAD + HW + c] = 0.0f;
  }

  v8f zero = {};
  v8f acc[7];
  #pragma unroll
  for (int i = 0; i < 7; ++i) acc[i] = zero;

  auto stage = [&](int s, int bufi) {
    const float* srcB = featb + s * (KS * HW);
    const float* srcA = conv_w + s * KS;
#if HAVE_ASYNC
    // feat slab: 16 rows x 55 x b128 chunks, direct global->LDS, no VGPR trip
    for (int idx = tid; idx < KS * (HW / 4); idx += 256) {
      int kk = idx / (HW / 4);
      int ch = idx - kk * (HW / 4);
      __builtin_amdgcn_global_load_async_to_lds_b128(
          (gv4p)(srcB + kk * HW + ch * 4),
          (lv4p)(&sB[bufi][kk * NPAD + ch * 4]), 0, 0);
    }
    // conv_w slab: 64 rows x 4 x b128 chunks = 256, exactly one per thread
    {
      int m  = tid >> 2;
      int ch = tid & 3;
      __builtin_amdgcn_global_load_async_to_lds_b128(
          (gv4p)(srcA + m * CB + ch * 4),
          (lv4p)(&sA[bufi][m * APAD + ch * 4]), 0, 0);
    }
#else
    for (int idx = tid; idx < KS * NPAD; idx += 256) {
      int kk = idx / NPAD;
      int hw = idx - kk * NPAD;
      if (hw < HW) sB[bufi][idx] = srcB[kk * HW + hw];
    }
    for (int idx = tid; idx < 64 * KS; idx += 256) {
      int m  = idx >> 4;          // KS == 16
      int kk = idx & (KS - 1);
      sA[bufi][m * APAD + kk] = srcA[m * CB + kk];
    }
#endif
  };

  stage(0, 0);
  WAIT_ASYNC();
  __syncthreads();

  for (int s = 0; s < CB / KS; ++s) {
    int cur = s & 1;
    if (s + 1 < CB / KS) stage(s + 1, cur ^ 1);   // overlap next slab loads

    const float* bA = &sA[cur][0];
    const float* bB = &sB[cur][0];
    #pragma unroll
    for (int kk = 0; kk < KS; kk += 4) {
      int kq = kk + (lh << 1);      // this lane's K pair base
      v2f afrag[4];                 // hoisted A frags, one per M-tile
      #pragma unroll
      for (int mt = 0; mt < 4; ++mt) {
        afrag[mt].x = bA[(mt * 16 + lr) * APAD + kq];
        afrag[mt].y = bA[(mt * 16 + lr) * APAD + kq + 1];
      }
      #pragma unroll
      for (int ti = 0; ti < 7; ++ti) {
        int t  = wv + ti * 8;       // tile 0..55
        int mt = t / 14;
        int nt = t - mt * 14;
        v2f bb;                     // B 4x16 frag: lane lr -> N, rows kq,kq+1
        bb.x = bB[kq * NPAD + nt * 16 + lr];
        bb.y = bB[(kq + 1) * NPAD + nt * 16 + lr];
        acc[ti] = __builtin_amdgcn_wmma_f32_16x16x4_f32(
            false, afrag[mt], false, bb, (short)0, acc[ti], false, false);
      }
    }
    WAIT_ASYNC();       // own async loads into next buffer landed
    __syncthreads();    // everyone done computing cur + loading next
  }

  // epilogue: bias + store f[b] = (64 x 220)
  for (int ti = 0; ti < 7; ++ti) {
    int t  = wv + ti * 8;
    int mt = t / 14;
    int nt = t - mt * 14;
    int n  = nt * 16 + lr;
    if (n < HW) {
      #pragma unroll
      for (int r = 0; r < 8; ++r) {
        int m = mt * 16 + r + (lh << 3);   // C/D layout: lanes 16-31 hold M+8
        fout[(size_t)b * (CF * HW) + m * HW + n] = acc[ti][r] + conv_b[m];
      }
    }
  }
}

// ---------------------------------------------------------------------------
// Kernel 2: anchor geometry + gather + attention + cls/reg head.
// 1 block per batch, 128 threads = 4 wave32s (one per proposal for attention).
// ---------------------------------------------------------------------------
__global__ __launch_bounds__(128) void laneatt_head(
    const float* __restrict__ fsrc, const float* __restrict__ rpn,
    const float* __restrict__ att_w, const float* __restrict__ att_b,
    const float* __restrict__ cls_w, const float* __restrict__ cls_b,
    const float* __restrict__ reg_w, const float* __restrict__ reg_b,
    float* __restrict__ out)
{
  __shared__ float af[NPROP][FDIM];     // anchor features
  __shared__ float attf[NPROP][FDIM];   // attention features
  __shared__ float attm[NPROP][NPROP];  // 4x4 attention matrix
  __shared__ float s_sx[NPROP], s_sy[NPROP], s_tn[NPROP];
  __shared__ int   s_cut[NPROP][FH];    // clamped cut index, -1 = invalid

  const int b   = blockIdx.x;
  const int tid = threadIdx.x;
  float* outb = out + (size_t)b * (NPROP * 77);

  // --- per-proposal anchor geometry ---
  if (tid < NPROP) {
    int p = tid;
    const float* rp = rpn + ((size_t)b * NPROP + p) * 4;
    float sx = rp[0] / IMG_W;
    float sy = rp[1] / IMG_H;
    float tn = tanf(rp[3] * 0.017453292519943295f);
    s_sx[p] = sx; s_sy[p] = sy; s_tn[p] = tn;
    for (int yy = 0; yy < FH; ++yy) {
      float acy = (float)yy * 0.1f;     // flipped anchor_cut_ys
      float xs  = (sx + (sy - acy) / tn) * IMG_W;
      int u  = (int)rintf(xs * (1.0f / 32.0f));          // round-half-even
      int cx = u < 0 ? 0 : (u > FW - 1 ? FW - 1 : u);
      s_cut[p][yy] = (u < 0 || u > FW) ? -1 : cx;        // faithful: > FW invalid
    }
    outb[p * 77 + 2] = 1.0f - sy;   // anchors[...,2]
    outb[p * 77 + 3] = sx;          // anchors[...,3]
  }
  __syncthreads();

  // --- gather anchor features: af[p][c*11+y] = f[b,c,y,cut] (0 if invalid) ---
  for (int idx = tid; idx < NPROP * FDIM; idx += 128) {
    int p = idx / FDIM;
    int r = idx - p * FDIM;
    int c = r / FH;
    int y = r - c * FH;
    int cx = s_cut[p][y];
    af[p][r] = (cx < 0) ? 0.0f
                        : fsrc[(size_t)b * (CF * HW) + c * HW + y * FW + cx];
  }
  __syncthreads();

  // --- attention scores + softmax: one wave per proposal ---
  {
    int wvp = tid >> 5;
    int lnp = tid & 31;
    float s0 = 0.f, s1 = 0.f, s2 = 0.f;
    for (int r = lnp; r < FDIM; r += 32) {
      float v = af[wvp][r];
      s0 += att_w[0 * FDIM + r] * v;
      s1 += att_w[1 * FDIM + r] * v;
      s2 += att_w[2 * FDIM + r] * v;
    }
    #pragma unroll
    for (int off = 16; off > 0; off >>= 1) {
      s0 += __shfl_xor(s0, off, 32);
      s1 += __shfl_xor(s1, off, 32);
      s2 += __shfl_xor(s2, off, 32);
    }
    if (lnp == 0) {
      s0 += att_b[0]; s1 += att_b[1]; s2 += att_b[2];
      float m  = fmaxf(s0, fmaxf(s1, s2));
      float e0 = __expf(s0 - m), e1 = __expf(s1 - m), e2 = __expf(s2 - m);
      float inv = 1.0f / (e0 + e1 + e2);
      float a3[3] = {e0 * inv, e1 * inv, e2 * inv};
      attm[wvp][wvp] = 0.0f;
      #pragma unroll
      for (int jj = 0; jj < 3; ++jj) {
        int col = jj < wvp ? jj : jj + 1;   // off-diagonal scatter
        attm[wvp][col] = a3[jj];
      }
    }
  }
  __syncthreads();

  // --- attention features: attf[i] = sum_j attm[i][j] * af[j] ---
  for (int idx = tid; idx < NPROP * FDIM; idx += 128) {
    int i = idx / FDIM;
    int r = idx - i * FDIM;
    attf[i][r] = attm[i][0] * af[0][r] + attm[i][1] * af[1][r]
               + attm[i][2] * af[2][r] + attm[i][3] * af[3][r];
  }
  __syncthreads();

  // --- cls (2) + reg (73) heads on cat = [attf, af], fused anchor assembly ---
  for (int task = tid; task < NPROP * 75; task += 128) {
    int i = task / 75;
    int o = task - i * 75;
    const float* wrow = (o < 2) ? (cls_w + o * 2 * FDIM)
                                : (reg_w + (o - 2) * 2 * FDIM);
    float accv = 0.0f;
    for (int f = 0; f < FDIM; ++f) accv += wrow[f] * attf[i][f];
    for (int f = 0; f < FDIM; ++f) accv += wrow[FDIM + f] * af[i][f];
    if (o < 2) {
      outb[i * 77 + o] = accv + cls_b[o];                 // cls_logits
    } else {
      int r = o - 2;
      float add = reg_b[r];
      if (r > 0) {   // anchors[...,5+s] = xs_full[s], s = r-1
        float ay = 1.0f - (float)(r - 1) * (1.0f / 71.0f);
        add += (s_sx[i] + (s_sy[i] - ay) / s_tn[i]) * IMG_W;
      }
      outb[i * 77 + 4 + r] = accv + add;
    }
  }
}

extern "C" void kernel_launch(void* const* d_in, const int* in_sizes, int n_in,
                              void* d_out, int out_size, void* d_ws, size_t ws_size,
                              hipStream_t stream) {
  const float* feat   = (const float*)d_in[0];
  const float* rpn    = (const float*)d_in[1];
  const float* conv_w = (const float*)d_in[2];
  const float* conv_b = (const float*)d_in[3];
  const float* att_w  = (const float*)d_in[4];
  const float* att_b  = (const float*)d_in[5];
  const float* cls_w  = (const float*)d_in[6];
  const float* cls_b  = (const float*)d_in[7];
  const float* reg_w  = (const float*)d_in[8];
  const float* reg_b  = (const float*)d_in[9];
  float* out  = (float*)d_out;
  float* fbuf = (float*)d_ws;              // B*64*220 floats = 28.8 MB scratch

  int B = in_sizes[1] / (NPROP * 4);       // 512

  laneatt_conv_wmma<<<B, 256, 0, stream>>>(feat, conv_w, conv_b, fbuf);
  laneatt_head<<<B, 128, 0, stream>>>(fbuf, rpn, att_w, att_b,
                                      cls_w, cls_b, reg_w, reg_b, out);
}